// half_integer_2bit_87703232184564
// MI455X (gfx1250) — compile-verified
//
#include <hip/hip_runtime.h>
#include <stdint.h>

// 1-D nearest-neighbor quantizer onto {-1.5,-0.5,0.5,1.5}.
// Pure streaming problem: 288 MiB of traffic, ~12 us floor at 23.3 TB/s HBM.
// Strategy: wave32 + b128 NT loads/stores, packed u8x4 index stores, prefetch.

typedef __attribute__((ext_vector_type(4))) float v4f;

__device__ __forceinline__ int qidx(float x) {
    // argmax tie-break (lower index wins) == strict '>' thresholds at -1,0,1
    return (int)(x > -1.0f) + (int)(x > 0.0f) + (int)(x > 1.0f);
}

__global__ __launch_bounds__(256)
void quantize_nn_b128(const v4f* __restrict__ x4,
                      v4f* __restrict__ xq4,
                      uint32_t* __restrict__ qi4,
                      int nvec, int n)
{
    const int gid = blockIdx.x * blockDim.x + threadIdx.x;

    if (gid < nvec) {
        // Pull the stream ahead into GL2 (emits global_prefetch_b8 on gfx1250).
        __builtin_prefetch((const char*)(x4 + gid) + 65536, 0, 0);

        // Read-once input: non-temporal b128 load.
        v4f v = __builtin_nontemporal_load(x4 + gid);

        const int i0 = qidx(v.x);
        const int i1 = qidx(v.y);
        const int i2 = qidx(v.z);
        const int i3 = qidx(v.w);

        // grid[i] = i - 1.5
        v4f q;
        q.x = (float)i0 - 1.5f;
        q.y = (float)i1 - 1.5f;
        q.z = (float)i2 - 1.5f;
        q.w = (float)i3 - 1.5f;

        // Write-once outputs: non-temporal stores, b128 for Xq, b32 packed u8x4 for idx.
        __builtin_nontemporal_store(q, xq4 + gid);

        const uint32_t packed = (uint32_t)i0 | ((uint32_t)i1 << 8) |
                                ((uint32_t)i2 << 16) | ((uint32_t)i3 << 24);
        __builtin_nontemporal_store(packed, qi4 + gid);
    }

    // Scalar tail for n not divisible by 4 (not hit for 8192*4096, kept for safety).
    if (gid == 0) {
        const float* x = (const float*)x4;
        float* xq = (float*)xq4;
        uint8_t* qi = (uint8_t*)qi4;
        for (int e = nvec * 4; e < n; ++e) {
            const int i = qidx(x[e]);
            xq[e] = (float)i - 1.5f;
            qi[e] = (uint8_t)i;
        }
    }
}

extern "C" void kernel_launch(void* const* d_in, const int* in_sizes, int n_in,
                              void* d_out, int out_size, void* d_ws, size_t ws_size,
                              hipStream_t stream) {
    (void)n_in; (void)out_size; (void)d_ws; (void)ws_size;

    const float* X = (const float*)d_in[0];   // [8192*4096*1] f32
    // d_in[1] = grid [4,1], d_in[2] = grid_norm [4]: compile-time constants of
    // the reference (GRID_VALS), folded into the kernel arithmetic.

    const int n    = in_sizes[0];
    const int nvec = n >> 2;

    // Output layout: Xq as n floats, then Xqidx as n uint8 bytes.
    float*    xq = (float*)d_out;
    uint32_t* qi = (uint32_t*)(xq + n);

    const int threads = 256;
    const int blocks  = (nvec + threads - 1) / threads;

    quantize_nn_b128<<<blocks, threads, 0, stream>>>(
        (const v4f*)X, (v4f*)xq, qi, nvec, n);
}